// CrossEntropyLoss_28467043238067
// MI455X (gfx1250) — compile-verified
//
#include <hip/hip_runtime.h>

typedef __attribute__((ext_vector_type(2))) float v2f;
typedef __attribute__((ext_vector_type(8))) float v8f;

#define B_ROWS  8192
#define VOCAB   32000
#define V4      (VOCAB / 4)      // 8000 float4 per row, 16B-aligned (row stride 128000 B)
#define THREADS 512              // 16 waves (wave32)
#define WAVES   (THREADS / 32)

// One workgroup per row: pass 1 = row max, pass 2 = sum of exp2((x-M)*log2e).
// Wave-level partial sums are reduced with V_WMMA_F32_16X16X4_F32 against a
// ones B-matrix (fp32-exact row-sum reduction on the matrix pipe).
__global__ __launch_bounds__(THREADS)
void ce_row_kernel(const float* __restrict__ outputs,
                   const long long* __restrict__ targets,
                   float* __restrict__ row_loss) {
    const int row = blockIdx.x;
    const int tid = threadIdx.x;
    const float* __restrict__ rp = outputs + (size_t)row * VOCAB;
    const float4* __restrict__ rp4 = (const float4*)rp;

    __shared__ float red[THREADS];
    __shared__ float wsum[2 * WAVES];

    // ---------------- pass 1: block max (streams row from HBM into L2) -----
    float m = -3.402823466e+38f;
    for (int i = tid; i < V4; i += THREADS) {
        float4 x = rp4[i];
        __builtin_prefetch(&rp4[i + THREADS], 0, 1);   // global_prefetch_b8
        m = fmaxf(m, fmaxf(fmaxf(x.x, x.y), fmaxf(x.z, x.w)));
    }
    red[tid] = m;
    __syncthreads();
    for (int s = THREADS / 2; s > 0; s >>= 1) {
        if (tid < s) red[tid] = fmaxf(red[tid], red[tid + s]);
        __syncthreads();
    }
    const float M = red[0];   // uniform across block

    // ---------------- pass 2: sum exp (re-read hits 192MB L2) --------------
    const float L2E = 1.44269504088896340736f;
    const float mb  = M * L2E;
    v2f acc = {0.0f, 0.0f};
    for (int i = tid; i < V4; i += THREADS) {
        float4 x = rp4[i];
        float e0 = __builtin_amdgcn_exp2f(__builtin_fmaf(x.x, L2E, -mb));
        float e1 = __builtin_amdgcn_exp2f(__builtin_fmaf(x.y, L2E, -mb));
        float e2 = __builtin_amdgcn_exp2f(__builtin_fmaf(x.z, L2E, -mb));
        float e3 = __builtin_amdgcn_exp2f(__builtin_fmaf(x.w, L2E, -mb));
        acc.x += e0 + e2;
        acc.y += e1 + e3;
    }

    // ---- cross-lane reduction via WMMA: D = A(16x4) * ones(4x16) ----------
    // A layout: lanes 0-15 carry K=0,1 (acc.x, acc.y); lanes 16-31 carry K=2,3.
    // With B = ones, D[m,n] = acc[m].x + acc[m].y + acc[m+16].x + acc[m+16].y.
    // Lane 0 holds D[0..7,0] and lane 16 holds D[8..15,0]; their C-sums add to
    // the exact fp32 wave total. EXEC is all-ones here (no divergence yet).
    v2f onesb = {1.0f, 1.0f};
    v8f c = {0.f, 0.f, 0.f, 0.f, 0.f, 0.f, 0.f, 0.f};
    c = __builtin_amdgcn_wmma_f32_16x16x4_f32(
            /*neg_a=*/false, acc, /*neg_b=*/false, onesb,
            /*c_mod=*/(short)0, c, /*reuse_a=*/false, /*reuse_b=*/false);
    float csum = c[0] + c[1] + c[2] + c[3] + c[4] + c[5] + c[6] + c[7];

    const int lane = tid & 31;
    const int wave = tid >> 5;
    if (lane == 0)  wsum[2 * wave + 0] = csum;
    if (lane == 16) wsum[2 * wave + 1] = csum;
    __syncthreads();

    if (tid == 0) {
        float S = 0.0f;
        for (int i = 0; i < 2 * WAVES; ++i) S += wsum[i];   // fixed order
        const long long t = targets[row];
        const float xt = rp[(int)t];
        // -logp[target] = M + ln(S) - x_t ;  ln(S) = ln2 * log2(S)
        const float loss = M + 0.69314718055994530942f * __builtin_amdgcn_logf(S) - xt;
        row_loss[row] = loss;
    }
}

// Deterministic fixed-order mean over the 8192 per-row losses.
__global__ __launch_bounds__(256)
void ce_mean_kernel(const float* __restrict__ row_loss, float* __restrict__ out) {
    __shared__ float red[256];
    float s = 0.0f;
    for (int i = threadIdx.x; i < B_ROWS; i += 256) s += row_loss[i];
    red[threadIdx.x] = s;
    __syncthreads();
    for (int t = 128; t > 0; t >>= 1) {
        if (threadIdx.x < t) red[threadIdx.x] += red[threadIdx.x + t];
        __syncthreads();
    }
    if (threadIdx.x == 0) out[0] = red[0] * (1.0f / (float)B_ROWS);
}

extern "C" void kernel_launch(void* const* d_in, const int* in_sizes, int n_in,
                              void* d_out, int out_size, void* d_ws, size_t ws_size,
                              hipStream_t stream) {
    const float*     outputs = (const float*)d_in[0];       // [8192, 32000] f32
    const long long* targets = (const long long*)d_in[1];   // [8192] i64
    float* row_loss = (float*)d_ws;                         // 8192 f32 scratch
    float* out      = (float*)d_out;                        // scalar f32

    ce_row_kernel<<<B_ROWS, THREADS, 0, stream>>>(outputs, targets, row_loss);
    ce_mean_kernel<<<1, 256, 0, stream>>>(row_loss, out);
}